// Messnode_39453569581174
// MI455X (gfx1250) — compile-verified
//
#include <hip/hip_runtime.h>
#include <stdint.h>

// Problem constants (reference: N=8192, D=256, K=32).
constexpr int N_ROWS = 8192;
constexpr int DCOLS  = 256;
constexpr int KSEL   = 32;
constexpr int TPB    = 256;                        // 8 waves (wave32)
constexpr int EPT    = N_ROWS / TPB;               // 32 keys per thread
constexpr int CHUNKS = (N_ROWS * 4) / (TPB * 16);  // 8 x b128 async loads per thread
constexpr int NWAVES = TPB / 32;

// Monotonic float->uint mapping: a > b (float) <=> f2ord(a) > f2ord(b) (uint).
__device__ __forceinline__ uint32_t f2ord(uint32_t u) {
  return (u & 0x80000000u) ? ~u : (u | 0x80000000u);
}

__global__ __launch_bounds__(TPB)
void topk_agg_kernel(const float* __restrict__ A,
                     const float* __restrict__ X,
                     const float* __restrict__ alphaP,
                     float* __restrict__ out)
{
  __shared__ __align__(16) uint32_t s_vals[N_ROWS];  // raw bits of A[row,:]
  __shared__ uint32_t s_hist[256];
  __shared__ uint32_t s_wsum[NWAVES];
  __shared__ uint32_t s_idx[KSEL];
  __shared__ uint32_t s_bc[2];
  __shared__ int s_pos, s_tie;

  const int tid  = threadIdx.x;
  const int lane = tid & 31;
  const int wid  = tid >> 5;
  const int row  = blockIdx.x;

  if (tid == 0) { s_pos = 0; s_tie = 0; }

  // ---- Phase 1: stream A[row,:] (32 KB) global -> LDS via async copy.
  // Non-temporal so the 256 MB A stream does not evict X (8 MB) from L2.
  {
    const uint64_t gbase = (uint64_t)(uintptr_t)(A + (size_t)row * N_ROWS);
    const uint32_t lbase = (uint32_t)(uintptr_t)(&s_vals[0]);  // low 32 bits = LDS offset
    #pragma unroll
    for (int c = 0; c < CHUNKS; ++c) {
      uint32_t loff  = lbase + (uint32_t)((c * TPB + tid) * 16);
      uint64_t gaddr = gbase + (uint64_t)((c * TPB + tid) * 16);
      asm volatile("global_load_async_to_lds_b128 %0, %1, off th:TH_LOAD_NT"
                   :: "v"(loff), "v"(gaddr) : "memory");
    }
    asm volatile("s_wait_asynccnt 0x0" ::: "memory");
  }
  __syncthreads();

  // ---- Phase 2: pull this thread's 32 keys into registers (stride-256
  // interleave -> consecutive lanes hit consecutive LDS banks).
  uint32_t kreg[EPT];
  #pragma unroll
  for (int i = 0; i < EPT; ++i)
    kreg[i] = f2ord(s_vals[i * TPB + tid]);

  // ---- Phase 3: exact radix-select of the K-th largest key (4 x 8-bit digits).
  uint32_t pref = 0;   // digits chosen so far (MSB first)
  uint32_t kth  = KSEL;
  for (int pass = 0; pass < 4; ++pass) {
    const int shift = 24 - 8 * pass;
    s_hist[tid] = 0;
    __syncthreads();
    #pragma unroll
    for (int i = 0; i < EPT; ++i) {
      uint32_t key = kreg[i];
      bool inClass = (pass == 0) || ((key >> (shift + 8)) == pref);
      if (inClass) atomicAdd(&s_hist[(key >> shift) & 255u], 1u);
    }
    __syncthreads();
    // Suffix count over bins: tid 0 -> bin 255, so an inclusive scan over tid
    // gives S = #elements-in-class with digit >= bin.
    const uint32_t bin = 255u - (uint32_t)tid;
    const uint32_t h   = s_hist[bin];
    uint32_t S = h;
    #pragma unroll
    for (int o = 1; o < 32; o <<= 1) {
      uint32_t n = __shfl_up(S, (unsigned)o, 32);
      if (lane >= o) S += n;
    }
    if (lane == 31) s_wsum[wid] = S;
    __syncthreads();
    uint32_t wpre = 0;
    #pragma unroll
    for (int w = 0; w < NWAVES; ++w)
      if (w < wid) wpre += s_wsum[w];
    S += wpre;
    // Exactly one thread owns the crossing bin.
    if (S >= kth && (S - h) < kth) { s_bc[0] = bin; s_bc[1] = S - h; }
    __syncthreads();
    pref = (pref << 8) | s_bc[0];
    kth -= s_bc[1];
    __syncthreads();
  }
  const uint32_t T    = pref;  // key value of the K-th largest element
  const uint32_t need = kth;   // how many ties at T still to take

  // ---- Phase 4: collect the K selected column indices.
  #pragma unroll
  for (int i = 0; i < EPT; ++i) {
    uint32_t key = kreg[i];
    if (key > T) {
      int p = atomicAdd(&s_pos, 1);
      s_idx[p] = (uint32_t)(i * TPB + tid);
    } else if (key == T) {
      int r = atomicAdd(&s_tie, 1);
      if ((uint32_t)r < need) {
        int p = atomicAdd(&s_pos, 1);
        s_idx[p] = (uint32_t)(i * TPB + tid);
      }
    }
  }
  __syncthreads();

  // ---- Phase 5: out[row,d] = X[row,d] + alpha * sum_j X[idx_j, d].
  // Thread t owns column t; X rows are 1 KB coalesced loads (L2 resident).
  const float alpha = alphaP[0];
  float acc = 0.0f;
  #pragma unroll
  for (int s = 0; s < KSEL; ++s)
    acc += X[(size_t)s_idx[s] * DCOLS + tid];
  out[(size_t)row * DCOLS + tid] = X[(size_t)row * DCOLS + tid] + alpha * acc;
}

extern "C" void kernel_launch(void* const* d_in, const int* in_sizes, int n_in,
                              void* d_out, int out_size, void* d_ws, size_t ws_size,
                              hipStream_t stream) {
  (void)in_sizes; (void)n_in; (void)d_ws; (void)ws_size; (void)out_size;
  const float* A     = (const float*)d_in[0];
  const float* X     = (const float*)d_in[1];
  const float* alpha = (const float*)d_in[2];
  // d_in[3] is k (== 32), compiled in as KSEL.
  float* out = (float*)d_out;
  topk_agg_kernel<<<dim3(N_ROWS), dim3(TPB), 0, stream>>>(A, X, alpha, out);
}